// Decoder_15247133901619
// MI455X (gfx1250) — compile-verified
//
#include <hip/hip_runtime.h>
#include <hip/hip_bf16.h>
#include <math.h>

// ---------------------------------------------------------------------------
// Decoder step: Bahdanau attention + 2-layer LSTM + vocab projection (fp32)
// V=32000 E=512 H=1024 ENC=1024 L=2 B=64 S=256
// Matrix math: CDNA5 V_WMMA_F32_16X16X4_F32 (wave32).
// Each wave owns one 16-col weight tile and computes ALL 64 batch rows
// (4 accumulators), so every weight byte is fetched from HBM exactly once:
// 1 weight b64 load feeds 4 WMMAs per K-step.
// ---------------------------------------------------------------------------

typedef float v2f __attribute__((ext_vector_type(2)));
typedef float v8f __attribute__((ext_vector_type(8)));

#define B_   64
#define S_   256
#define H_   1024
#define ENC_ 1024
#define E_   512
#define V_   32000
#define XK_  (E_ + ENC_)        // 1536 : LSTM0 input width
#define FK_  (H_ + ENC_ + E_)   // 2560 : output-projection feature width
#define G4H_ (4 * H_)           // 4096 : gate width

__device__ __forceinline__ float sigmoidf_(float x) {
    return 1.0f / (1.0f + __expf(-x));
}

// acc[r] += Xtile_r(16x4) * W^T tile(4x16), r = 0..3 (rows 16r..16r+15).
// Fragment layout (ISA 7.12.2, 32-bit 16x4 A / 16x16 C):
//   lr = lane&15, hi = lane>>4
//   A: lane holds X[16r+lr][k+2*hi], X[16r+lr][k+2*hi+1]
//   B: lane holds W[col0+lr][k+2*hi], W[col0+lr][k+2*hi+1]   (B = W^T)
//   C: vgpr j holds C[16r + j + 8*hi][col0 + lr]
__device__ __forceinline__ void wmma_reduce_m64(const float* __restrict__ X,
                                                const float* __restrict__ W,
                                                int K, int col0, int lr, int hi,
                                                v8f acc[4]) {
    const float* xr = X + (size_t)lr * K + 2 * hi;
    const float* wr = W + (size_t)(col0 + lr) * K + 2 * hi;
#pragma unroll 2
    for (int k = 0; k < K; k += 4) {
        const v2f bfrag = *(const v2f*)(wr + k);   // one weight fetch ...
#pragma unroll
        for (int r = 0; r < 4; ++r) {              // ... feeds 4 WMMAs
            const v2f afrag = *(const v2f*)(xr + (size_t)(16 * r) * K + k);
            acc[r] = __builtin_amdgcn_wmma_f32_16x16x4_f32(
                /*neg_a=*/false, afrag, /*neg_b=*/false, bfrag,
                /*c_mod=*/(short)0, acc[r], /*reuse_a=*/false, /*reuse_b=*/false);
        }
    }
}

// out[64,N] = X[64,K] @ W[N,K]^T (+ X2[64,K2] @ W2[N,K2]^T) (+ bias[N])
// grid = N/128, block = 256 (8 waves; each wave: 16 cols x all 64 rows).
__global__ __launch_bounds__(256)
void gemm_xwT_m64_kernel(const float* __restrict__ X,  const float* __restrict__ W,  int K,
                         const float* __restrict__ X2, const float* __restrict__ W2, int K2,
                         const float* __restrict__ bias, float* __restrict__ out, int N) {
    const int wave = threadIdx.x >> 5;
    const int lane = threadIdx.x & 31;
    const int lr   = lane & 15;
    const int hi   = lane >> 4;
    const int col0 = blockIdx.x * 128 + wave * 16;

    v8f acc[4] = {};
    wmma_reduce_m64(X, W, K, col0, lr, hi, acc);
    if (X2 != nullptr) {  // wave-uniform branch: EXEC stays all-ones for WMMA
        wmma_reduce_m64(X2, W2, K2, col0, lr, hi, acc);
    }
    const float bv = bias ? bias[col0 + lr] : 0.0f;
#pragma unroll
    for (int r = 0; r < 4; ++r) {
#pragma unroll
        for (int j = 0; j < 8; ++j) {
            out[(size_t)(16 * r + j + 8 * hi) * N + (col0 + lr)] = acc[r][j] + bv;
        }
    }
}

// embedded = embedding[token]; scatter into x0[:, :E] and feat[:, H+ENC:].
__global__ __launch_bounds__(256)
void embed_gather_kernel(const int* __restrict__ tok,
                         const float* __restrict__ embedding,
                         float* __restrict__ x0, float* __restrict__ feat) {
    const int b = blockIdx.x;
    const int t = threadIdx.x;
    const size_t row = (size_t)tok[b] * E_;
#pragma unroll
    for (int i = 0; i < E_ / 256; ++i) {
        const int e = t + 256 * i;
        const float val = embedding[row + e];
        x0[(size_t)b * XK_ + e]                 = val;
        feat[(size_t)b * FK_ + (H_ + ENC_) + e] = val;
    }
}

// Fused keys-GEMM + additive-attention energy (keys never materialized).
// energy[bs] = sum_h v[h] * tanh(query[b,h] + (enc[bs,:] @ Wk[h,:]))  (masked)
// grid = B*S/64 blocks; block = 128 (4 waves). Each block owns 64 bs rows
// (same batch b since 64 | S); wave w handles h-tiles w, w+4, ... and each
// K-step shares one Wk fragment across 4 row-tile WMMAs. Per-row partial
// energies from the 4 waves are summed via LDS.
__global__ __launch_bounds__(128)
void attn_energy_kernel(const float* __restrict__ enc,    // [B*S, ENC]
                        const float* __restrict__ Wk,     // [H, ENC]
                        const float* __restrict__ query,  // [B, H]
                        const float* __restrict__ vvec,   // [H]
                        const int*   __restrict__ mask,   // [B*S]
                        float* __restrict__ energy) {     // [B*S]
    const int wave = threadIdx.x >> 5;
    const int lane = threadIdx.x & 31;
    const int lr   = lane & 15;
    const int hi   = lane >> 4;
    const int row0 = blockIdx.x * 64;
    const int b    = row0 / S_;

    __shared__ float red[4][64];

    float partial[4][8];
#pragma unroll
    for (int r = 0; r < 4; ++r)
#pragma unroll
        for (int j = 0; j < 8; ++j) partial[r][j] = 0.0f;

    for (int hc = wave; hc < H_ / 16; hc += 4) {
        const int h0 = hc * 16;
        v8f acc[4] = {};
        const float* xr = enc + (size_t)(row0 + lr) * ENC_ + 2 * hi;
        const float* wr = Wk + (size_t)(h0 + lr) * ENC_ + 2 * hi;
#pragma unroll 2
        for (int k = 0; k < ENC_; k += 4) {
            const v2f bfrag = *(const v2f*)(wr + k);
#pragma unroll
            for (int r = 0; r < 4; ++r) {
                const v2f afrag = *(const v2f*)(xr + (size_t)(16 * r) * ENC_ + k);
                acc[r] = __builtin_amdgcn_wmma_f32_16x16x4_f32(
                    false, afrag, false, bfrag, (short)0, acc[r], false, false);
            }
        }
        const float vv = vvec[h0 + lr];
        const float qq = query[(size_t)b * H_ + h0 + lr];
#pragma unroll
        for (int r = 0; r < 4; ++r)
#pragma unroll
            for (int j = 0; j < 8; ++j)
                partial[r][j] += vv * tanhf(qq + acc[r][j]);
    }
    // Reduce over the 16 h-columns held by lanes within each half-wave.
#pragma unroll
    for (int r = 0; r < 4; ++r) {
#pragma unroll
        for (int j = 0; j < 8; ++j) {
            float p = partial[r][j];
            p += __shfl_xor(p, 1, 16);
            p += __shfl_xor(p, 2, 16);
            p += __shfl_xor(p, 4, 16);
            p += __shfl_xor(p, 8, 16);
            partial[r][j] = p;
        }
    }
    if (lr == 0) {
#pragma unroll
        for (int r = 0; r < 4; ++r)
#pragma unroll
            for (int j = 0; j < 8; ++j)
                red[wave][16 * r + 8 * hi + j] = partial[r][j];
    }
    __syncthreads();
    // Cross-wave sum + mask + store (threads 0..63, one per bs row).
    if (threadIdx.x < 64) {
        const int rrow = threadIdx.x;
        float e = red[0][rrow] + red[1][rrow] + red[2][rrow] + red[3][rrow];
        const int bs = row0 + rrow;
        if (mask[bs] == 0) e = -1e10f;
        energy[bs] = e;
    }
}

// Softmax over S per batch row. grid = B, block = S = 256.
__global__ __launch_bounds__(256)
void softmax_kernel(const float* __restrict__ energy, float* __restrict__ attn) {
    __shared__ float red[256];
    const int b = blockIdx.x;
    const int t = threadIdx.x;
    const float e = energy[(size_t)b * S_ + t];

    red[t] = e;
    __syncthreads();
    for (int o = 128; o > 0; o >>= 1) {
        if (t < o) red[t] = fmaxf(red[t], red[t + o]);
        __syncthreads();
    }
    const float m = red[0];
    __syncthreads();

    const float ex = __expf(e - m);
    red[t] = ex;
    __syncthreads();
    for (int o = 128; o > 0; o >>= 1) {
        if (t < o) red[t] += red[t + o];
        __syncthreads();
    }
    attn[(size_t)b * S_ + t] = ex / red[0];
}

// context[b,:] = sum_s attn[b,s] * enc[b,s,:]; scatter into x0[:,E:] and
// feat[:,H:H+ENC]. grid = B, block = 256 (each thread: 4 strided columns).
__global__ __launch_bounds__(256)
void context_kernel(const float* __restrict__ attn, const float* __restrict__ enc,
                    float* __restrict__ x0, float* __restrict__ feat) {
    const int b = blockIdx.x;
    const int t = threadIdx.x;
    float acc[4] = {0.f, 0.f, 0.f, 0.f};
    const float* arow = attn + (size_t)b * S_;
    const float* erow = enc + (size_t)b * S_ * ENC_;
    for (int s = 0; s < S_; ++s) {
        const float a = arow[s];
        const float* r = erow + (size_t)s * ENC_;
#pragma unroll
        for (int i = 0; i < 4; ++i) acc[i] += a * r[t + 256 * i];
    }
#pragma unroll
    for (int i = 0; i < 4; ++i) {
        const int e = t + 256 * i;
        x0[(size_t)b * XK_ + E_ + e]   = acc[i];
        feat[(size_t)b * FK_ + H_ + e] = acc[i];
    }
}

// Elementwise LSTM cell from precomputed gates (PyTorch order i,f,g,o).
__global__ __launch_bounds__(256)
void lstm_elem_kernel(const float* __restrict__ gates,   // [B, 4H]
                      const float* __restrict__ bih, const float* __restrict__ bhh,
                      const float* __restrict__ c_prev,  // [B*H]
                      float* __restrict__ h_out, float* __restrict__ c_out,
                      float* __restrict__ h_out2, int ld2, int off2) {
    const int idx = blockIdx.x * blockDim.x + threadIdx.x;  // 0 .. B*H-1
    const int b = idx / H_;
    const int h = idx % H_;
    const float* g = gates + (size_t)b * G4H_;
    const float gi = g[h]          + bih[h]          + bhh[h];
    const float gf = g[H_ + h]     + bih[H_ + h]     + bhh[H_ + h];
    const float gg = g[2 * H_ + h] + bih[2 * H_ + h] + bhh[2 * H_ + h];
    const float go = g[3 * H_ + h] + bih[3 * H_ + h] + bhh[3 * H_ + h];
    const float c  = sigmoidf_(gf) * c_prev[idx] + sigmoidf_(gi) * tanhf(gg);
    const float hn = sigmoidf_(go) * tanhf(c);
    c_out[idx] = c;
    h_out[idx] = hn;
    if (h_out2 != nullptr) h_out2[(size_t)b * ld2 + off2 + h] = hn;
}

extern "C" void kernel_launch(void* const* d_in, const int* in_sizes, int n_in,
                              void* d_out, int out_size, void* d_ws, size_t ws_size,
                              hipStream_t stream) {
    (void)in_sizes; (void)n_in; (void)out_size; (void)ws_size;

    const int*   tok    = (const int*)  d_in[0];
    const float* hidden = (const float*)d_in[1];   // [L,B,H]
    const float* cell   = (const float*)d_in[2];   // [L,B,H]
    const float* enc    = (const float*)d_in[3];   // [B,S,ENC]
    const int*   mask   = (const int*)  d_in[4];   // [B,S]
    const float* emb    = (const float*)d_in[5];   // [V,E]
    const float* Wq     = (const float*)d_in[6];   // [H,H]
    const float* Wk     = (const float*)d_in[7];   // [H,ENC]
    const float* vvec   = (const float*)d_in[8];   // [H]
    const float* Wih0   = (const float*)d_in[9];   // [4H, E+ENC]
    const float* Whh0   = (const float*)d_in[10];  // [4H, H]
    const float* bih0   = (const float*)d_in[11];
    const float* bhh0   = (const float*)d_in[12];
    const float* Wih1   = (const float*)d_in[13];  // [4H, H]
    const float* Whh1   = (const float*)d_in[14];  // [4H, H]
    const float* bih1   = (const float*)d_in[15];
    const float* bhh1   = (const float*)d_in[16];
    const float* Wout   = (const float*)d_in[17];  // [V, H+ENC+E]
    const float* bout   = (const float*)d_in[18];

    // d_out layout: prediction | new_hidden | new_cell | attn_weights
    float* pred = (float*)d_out;                         // B*V
    float* hid  = pred + (size_t)B_ * V_;                // L*B*H
    float* cel  = hid + (size_t)2 * B_ * H_;             // L*B*H
    float* attn = cel + (size_t)2 * B_ * H_;             // B*S

    // Workspace layout (floats), ~2.4 MB total.
    float* ws     = (float*)d_ws;
    float* query  = ws;                                  // B*H      = 65536
    float* energy = query + (size_t)B_ * H_;             // B*S      = 16384
    float* x0     = energy + (size_t)B_ * S_;            // B*XK     = 98304
    float* feat   = x0 + (size_t)B_ * XK_;               // B*FK     = 163840
    float* gates  = feat + (size_t)B_ * FK_;             // B*4H     = 262144

    const dim3 blk256(256);
    const dim3 blk128(128);

    // 1) embedding gather -> x0[:, :E], feat[:, H+ENC:]
    embed_gather_kernel<<<dim3(B_), blk256, 0, stream>>>(tok, emb, x0, feat);

    // 2) query = hidden[-1] @ Wq^T           [64,1024]x[1024,1024]
    gemm_xwT_m64_kernel<<<dim3(H_ / 128), blk256, 0, stream>>>(
        hidden + (size_t)1 * B_ * H_, Wq, H_, nullptr, nullptr, 0, nullptr, query, H_);

    // 3) fused keys GEMM + tanh + v-dot -> energy (masked)
    attn_energy_kernel<<<dim3((B_ * S_) / 64), blk128, 0, stream>>>(
        enc, Wk, query, vvec, mask, energy);

    // 4) softmax over S -> attn (directly into d_out slot)
    softmax_kernel<<<dim3(B_), blk256, 0, stream>>>(energy, attn);

    // 5) context -> x0[:, E:], feat[:, H:H+ENC]
    context_kernel<<<dim3(B_), blk256, 0, stream>>>(attn, enc, x0, feat);

    // 6) layer-0 gates = x0 @ Wih0^T + h_prev0 @ Whh0^T   [64,4096]
    gemm_xwT_m64_kernel<<<dim3(G4H_ / 128), blk256, 0, stream>>>(
        x0, Wih0, XK_, hidden, Whh0, H_, nullptr, gates, G4H_);
    lstm_elem_kernel<<<dim3((B_ * H_) / 256), blk256, 0, stream>>>(
        gates, bih0, bhh0, cell, hid, cel, nullptr, 0, 0);

    // 7) layer-1 gates = h0_new @ Wih1^T + h_prev1 @ Whh1^T
    gemm_xwT_m64_kernel<<<dim3(G4H_ / 128), blk256, 0, stream>>>(
        hid, Wih1, H_, hidden + (size_t)B_ * H_, Whh1, H_, nullptr, gates, G4H_);
    lstm_elem_kernel<<<dim3((B_ * H_) / 256), blk256, 0, stream>>>(
        gates, bih1, bhh1, cell + (size_t)B_ * H_,
        hid + (size_t)B_ * H_, cel + (size_t)B_ * H_, feat, FK_, 0);

    // 8) prediction = feat @ Wout^T + bout   [64,2560]x[2560,32000]
    gemm_xwT_m64_kernel<<<dim3(V_ / 128), blk256, 0, stream>>>(
        feat, Wout, FK_, nullptr, nullptr, 0, bout, pred, V_);
}